// GPRMamba2_38190849196467
// MI455X (gfx1250) — compile-verified
//
#include <hip/hip_runtime.h>
#include <math.h>

typedef float v2f __attribute__((ext_vector_type(2)));
typedef float v8f __attribute__((ext_vector_type(8)));

#define DD     128
#define HH     16
#define WW     32
#define LL     512
#define CINCH  3
#define NHD    4
#define HDIM   64
#define NSTATE 128
#define INTERC 256
#define CONVC  512
#define PROJC  772
#define KCONV  4
#define NLAY   4
#define NSEQ   16
#define EPSV   1e-5f

#define WMMA_F32(A, B, C) \
  __builtin_amdgcn_wmma_f32_16x16x4_f32(false, (A), false, (B), (short)0, (C), false, false)

// ---------------------------------------------------------------------------
// patch embed + cross_scan: hs[16][512][128]
// ---------------------------------------------------------------------------
__global__ void patch_scan_kernel(const float* __restrict__ x,
                                  const float* __restrict__ pw,
                                  const float* __restrict__ pb,
                                  float* __restrict__ hs) {
  int i = blockIdx.x * blockDim.x + threadIdx.x;
  if (i >= NSEQ * LL * DD) return;
  int d = i % DD;
  int l = (i / DD) % LL;
  int s = i / (DD * LL);
  int dir = s >> 2, b = s & 3;
  int lsrc;
  if (dir == 0)      lsrc = l;
  else if (dir == 1) lsrc = LL - 1 - l;
  else {
    int le = (dir == 2) ? l : (LL - 1 - l);     // d2 index (w*H + h)
    int w = le / HH, h = le % HH;
    lsrc = h * WW + w;                          // row-major source
  }
  float acc = pb[d];
#pragma unroll
  for (int c = 0; c < CINCH; c++)
    acc += x[(b * CINCH + c) * LL + lsrc] * pw[d * CINCH + c];
  hs[i] = acc;
}

// ---------------------------------------------------------------------------
// rmsnorm: one block per row
// ---------------------------------------------------------------------------
template <int WIDTH>
__global__ void rmsnorm_kernel(const float* __restrict__ in,
                               const float* __restrict__ w,
                               float* __restrict__ out) {
  __shared__ float sd[WIDTH];
  int row = blockIdx.x, tid = threadIdx.x;
  float v = in[(size_t)row * WIDTH + tid];
  sd[tid] = v * v;
  __syncthreads();
  for (int off = WIDTH / 2; off > 0; off >>= 1) {
    if (tid < off) sd[tid] += sd[tid + off];
    __syncthreads();
  }
  float scale = rsqrtf(sd[0] / (float)WIDTH + EPSV);
  out[(size_t)row * WIDTH + tid] = v * scale * w[tid];
}

// ---------------------------------------------------------------------------
// y = rmsnorm(y * silu(gate), gw), in place over y (width = INTERC)
// ---------------------------------------------------------------------------
__global__ void gated_rmsnorm_kernel(float* __restrict__ y,
                                     const float* __restrict__ proj,
                                     const float* __restrict__ gw) {
  __shared__ float sd[INTERC];
  int row = blockIdx.x, tid = threadIdx.x;
  float g = proj[(size_t)row * PROJC + tid];          // gate = proj[..., :INTER]
  float v = y[(size_t)row * INTERC + tid] * (g / (1.f + expf(-g)));
  sd[tid] = v * v;
  __syncthreads();
  for (int off = INTERC / 2; off > 0; off >>= 1) {
    if (tid < off) sd[tid] += sd[tid + off];
    __syncthreads();
  }
  float scale = rsqrtf(sd[0] / (float)INTERC + EPSV);
  y[(size_t)row * INTERC + tid] = v * scale * gw[tid];
}

// ---------------------------------------------------------------------------
// causal depthwise conv (K=4) + silu: proj[...,256:768] -> xc[16][512][512]
// ---------------------------------------------------------------------------
__global__ void conv_silu_kernel(const float* __restrict__ proj,
                                 const float* __restrict__ cw,
                                 const float* __restrict__ cb,
                                 float* __restrict__ xc) {
  int i = blockIdx.x * blockDim.x + threadIdx.x;
  if (i >= NSEQ * LL * CONVC) return;
  int c = i % CONVC;
  int l = (i / CONVC) % LL;
  int s = i / (CONVC * LL);
  const float* pr = proj + (size_t)(s * LL) * PROJC + INTERC + c;
  float acc = cb[c];
#pragma unroll
  for (int k = 0; k < KCONV; k++) {
    int ls = l + k - (KCONV - 1);
    if (ls >= 0) acc += pr[(size_t)ls * PROJC] * cw[c * KCONV + k];
  }
  xc[i] = acc / (1.f + expf(-acc));
}

// ---------------------------------------------------------------------------
// dt = softplus(proj[...,768:772]+dtb); s = cumsum(dt*A) per (seq,head)
// ---------------------------------------------------------------------------
__global__ void dtscan_kernel(const float* __restrict__ proj,
                              const float* __restrict__ dtbias,
                              const float* __restrict__ alog,
                              float* __restrict__ dtb,
                              float* __restrict__ sb) {
  __shared__ float sd[LL];
  int s = blockIdx.x >> 2, h = blockIdx.x & 3;
  int l = threadIdx.x;
  float raw = proj[(size_t)(s * LL + l) * PROJC + INTERC + CONVC + h] + dtbias[h];
  float dt = raw > 20.f ? raw : log1pf(expf(raw));
  dtb[(size_t)(s * LL + l) * NHD + h] = dt;
  float A = -expf(alog[h]);
  sd[l] = dt * A;
  __syncthreads();
  for (int off = 1; off < LL; off <<= 1) {
    float add = (l >= off) ? sd[l - off] : 0.f;
    __syncthreads();
    sd[l] += add;
    __syncthreads();
  }
  sb[(size_t)(s * NHD + h) * LL + l] = sd[l];
}

// ---------------------------------------------------------------------------
// Batched WMMA GEMM (fp32, 16x16x4): C[m,n] = sum_k A[m,k]*B[n,k]
// One wave computes 32(M) x 64(N): 8 accumulators, ping-pong software
// pipeline (two fragment sets, no cross-iteration register copies).
// No predication in K-loop: OOB columns CLAMPED (garbage cols never stored).
// Requires: M %32 == 0, K %8 == 0.
// flags: bit0 = accumulate into C, bit1 = exact GELU
// ---------------------------------------------------------------------------
#define GLOADF(aa0, aa1, bb0, bb1, bb2, bb3, kk)  \
  aa0 = *(const v2f*)(ap0 + (kk));                \
  aa1 = *(const v2f*)(ap1 + (kk));                \
  bb0 = *(const v2f*)(bp0 + (kk));                \
  bb1 = *(const v2f*)(bp1 + (kk));                \
  bb2 = *(const v2f*)(bp2 + (kk));                \
  bb3 = *(const v2f*)(bp3 + (kk));

#define GMMA8(aa0, aa1, bb0, bb1, bb2, bb3)       \
  acc00 = WMMA_F32(aa0, bb0, acc00);              \
  acc01 = WMMA_F32(aa0, bb1, acc01);              \
  acc02 = WMMA_F32(aa0, bb2, acc02);              \
  acc03 = WMMA_F32(aa0, bb3, acc03);              \
  acc10 = WMMA_F32(aa1, bb0, acc10);              \
  acc11 = WMMA_F32(aa1, bb1, acc11);              \
  acc12 = WMMA_F32(aa1, bb2, acc12);              \
  acc13 = WMMA_F32(aa1, bb3, acc13);

__global__ __launch_bounds__(128)
void wmma_gemm_kernel(const float* __restrict__ A, long sA, int lda,
                      const float* __restrict__ Bw, long sB, int ldb,
                      float* __restrict__ C, long sC, int ldc,
                      const float* __restrict__ bias,
                      int M, int Nn, int Kk, int flags) {
  int wave = threadIdx.x >> 5;
  int lane = threadIdx.x & 31;
  int tilesM2 = M >> 5;
  int tilesN4 = (Nn + 63) >> 6;
  int tile = blockIdx.x * (blockDim.x >> 5) + wave;
  if (tile >= tilesM2 * tilesN4) return;            // wave-uniform exit
  int tm = tile / tilesN4, tn4 = tile % tilesN4;
  const float* Ab = A + (size_t)blockIdx.y * sA;
  const float* Bb = Bw + (size_t)blockIdx.y * sB;
  float*       Cb = C + (size_t)blockIdx.y * sC;

  int idx = lane & 15, kh = lane >> 4;
  const float* ap0 = Ab + (size_t)(tm * 32 + idx) * lda;
  const float* ap1 = ap0 + (size_t)16 * lda;

  int n0 = tn4 * 64 + idx;
  int c0 = n0      < Nn ? n0      : Nn - 1;         // clamp, no predication
  int c1 = n0 + 16 < Nn ? n0 + 16 : Nn - 1;
  int c2 = n0 + 32 < Nn ? n0 + 32 : Nn - 1;
  int c3 = n0 + 48 < Nn ? n0 + 48 : Nn - 1;
  const float* bp0 = Bb + (size_t)c0 * ldb;
  const float* bp1 = Bb + (size_t)c1 * ldb;
  const float* bp2 = Bb + (size_t)c2 * ldb;
  const float* bp3 = Bb + (size_t)c3 * ldb;

  v8f z = {0.f,0.f,0.f,0.f,0.f,0.f,0.f,0.f};
  v8f acc00 = z, acc01 = z, acc02 = z, acc03 = z;
  v8f acc10 = z, acc11 = z, acc12 = z, acc13 = z;

  v2f a0A, a1A, b0A, b1A, b2A, b3A;
  v2f a0B, a1B, b0B, b1B, b2B, b3B;

  int k = 2 * kh;                                   // kbase
  GLOADF(a0A, a1A, b0A, b1A, b2A, b3A, k);  k += 4;
  int pairs = (Kk >> 3) - 1;                        // K %8 == 0
  for (int p = 0; p < pairs; p++) {
    GLOADF(a0B, a1B, b0B, b1B, b2B, b3B, k);  k += 4;
    GMMA8(a0A, a1A, b0A, b1A, b2A, b3A);
    GLOADF(a0A, a1A, b0A, b1A, b2A, b3A, k);  k += 4;
    GMMA8(a0B, a1B, b0B, b1B, b2B, b3B);
  }
  GLOADF(a0B, a1B, b0B, b1B, b2B, b3B, k);
  GMMA8(a0A, a1A, b0A, b1A, b2A, b3A);
  GMMA8(a0B, a1B, b0B, b1B, b2B, b3B);

  v8f accs[2][4] = {{acc00, acc01, acc02, acc03}, {acc10, acc11, acc12, acc13}};
#pragma unroll
  for (int mt = 0; mt < 2; mt++) {
#pragma unroll
    for (int t = 0; t < 4; t++) {
      int n = n0 + t * 16;
      if (n < Nn) {
        float bv = bias ? bias[n] : 0.f;
        v8f av = accs[mt][t];
#pragma unroll
        for (int j = 0; j < 8; j++) {
          int m = tm * 32 + mt * 16 + j + 8 * kh;
          float v = av[j] + bv;
          if (flags & 2) v = 0.5f * v * (1.f + erff(v * 0.70710678118654752f));
          size_t off = (size_t)m * ldc + n;
          if (flags & 1) v += Cb[off];
          Cb[off] = v;
        }
      }
    }
  }
}

// ---------------------------------------------------------------------------
// Decay attention: y[t,h,p] = sum_{k<=t} exp(s_t - s_k)*CB[t,k]*dt_k * xs[k,h,p]
//                 + D[h]*xs[t,h,p]
// One wave per 16(t) x 64(p=HDIM) block; exp-built A fragment reused over 4
// head-dim tiles; ping-pong pipeline over K (kend %16 == 0 so it divides
// evenly). Masking by value select only; EXEC stays full for WMMA.
// ---------------------------------------------------------------------------
#define ALOAD(skX, cbX, d0X, d1X, bb0, bb1, bb2, bb3, kk) {  \
  skX = *(const v2f*)(sbh + (kk));                           \
  cbX = *(const v2f*)(cbr + (kk));                           \
  d0X = dth[(size_t)(kk) * NHD];                             \
  d1X = dth[(size_t)((kk) + 1) * NHD];                       \
  const float* xr0_ = xch + (size_t)(kk) * CONVC;            \
  const float* xr1_ = xr0_ + CONVC;                          \
  bb0.x = xr0_[idx];      bb0.y = xr1_[idx];                 \
  bb1.x = xr0_[idx + 16]; bb1.y = xr1_[idx + 16];            \
  bb2.x = xr0_[idx + 32]; bb2.y = xr1_[idx + 32];            \
  bb3.x = xr0_[idx + 48]; bb3.y = xr1_[idx + 48]; }

#define AMMA4(skX, cbX, d0X, d1X, bb0, bb1, bb2, bb3, kaX) { \
  float m0_ = expf(st - skX.x) * cbX.x * d0X;                \
  float m1_ = expf(st - skX.y) * cbX.y * d1X;                \
  v2f a_;                                                    \
  a_.x = ((kaX)     <= t) ? m0_ : 0.f;                       \
  a_.y = ((kaX) + 1 <= t) ? m1_ : 0.f;                       \
  acc0 = WMMA_F32(a_, bb0, acc0);                            \
  acc1 = WMMA_F32(a_, bb1, acc1);                            \
  acc2 = WMMA_F32(a_, bb2, acc2);                            \
  acc3 = WMMA_F32(a_, bb3, acc3); }

__global__ __launch_bounds__(128)
void ssm_attn_kernel(const float* __restrict__ CBm,
                     const float* __restrict__ sb,
                     const float* __restrict__ dtb,
                     const float* __restrict__ xc,
                     const float* __restrict__ dpar,
                     float* __restrict__ y) {
  int s = blockIdx.y >> 2, h = blockIdx.y & 3;
  int wave = threadIdx.x >> 5;
  int lane = threadIdx.x & 31;
  int tt = blockIdx.x * 4 + wave;                   // 0..31 (t-tile)
  int idx = lane & 15, kh = lane >> 4;
  int t = tt * 16 + idx;                            // A row (time)
  const float* sbh = sb + (size_t)(s * NHD + h) * LL;
  const float* cbr = CBm + (size_t)s * LL * LL + (size_t)t * LL;
  const float* dth = dtb + (size_t)s * LL * NHD + h;
  const float* xch = xc + (size_t)s * LL * CONVC + h * HDIM;
  float st = sbh[t];

  v8f z = {0.f,0.f,0.f,0.f,0.f,0.f,0.f,0.f};
  v8f acc0 = z, acc1 = z, acc2 = z, acc3 = z;

  int kend = (tt + 1) * 16;                         // causal, wave-uniform
  v2f skA, cbA, b0A, b1A, b2A, b3A;
  v2f skB, cbB, b0B, b1B, b2B, b3B;
  float d0A, d1A, d0B, d1B;

  int k = 2 * kh;                                   // kbase
  int kA, kB;
  ALOAD(skA, cbA, d0A, d1A, b0A, b1A, b2A, b3A, k); kA = k; k += 4;
  int pairs = (kend >> 3) - 1;                      // kend %16 == 0 -> >= 1
  for (int p = 0; p < pairs; p++) {
    ALOAD(skB, cbB, d0B, d1B, b0B, b1B, b2B, b3B, k); kB = k; k += 4;
    AMMA4(skA, cbA, d0A, d1A, b0A, b1A, b2A, b3A, kA);
    ALOAD(skA, cbA, d0A, d1A, b0A, b1A, b2A, b3A, k); kA = k; k += 4;
    AMMA4(skB, cbB, d0B, d1B, b0B, b1B, b2B, b3B, kB);
  }
  ALOAD(skB, cbB, d0B, d1B, b0B, b1B, b2B, b3B, k); kB = k;
  AMMA4(skA, cbA, d0A, d1A, b0A, b1A, b2A, b3A, kA);
  AMMA4(skB, cbB, d0B, d1B, b0B, b1B, b2B, b3B, kB);

  float dp = dpar[h];
  v8f accs[4] = {acc0, acc1, acc2, acc3};
#pragma unroll
  for (int tp = 0; tp < 4; tp++) {
    int p = tp * 16 + idx;
    v8f av = accs[tp];
#pragma unroll
    for (int j = 0; j < 8; j++) {
      int to = tt * 16 + j + 8 * kh;
      float v = av[j] + dp * xch[(size_t)to * CONVC + p];
      y[(size_t)(s * LL + to) * INTERC + h * HDIM + p] = v;
    }
  }
}

// ---------------------------------------------------------------------------
// cross_merge: m[b][l][d] from u[16][512][128]
// ---------------------------------------------------------------------------
__global__ void merge_kernel(const float* __restrict__ u, float* __restrict__ m) {
  int i = blockIdx.x * blockDim.x + threadIdx.x;
  if (i >= 4 * LL * DD) return;
  int d = i % DD;
  int l = (i / DD) % LL;
  int b = i / (DD * LL);
  int hcoord = l / WW, wcoord = l % WW;
  int lt = wcoord * HH + hcoord;                    // transposed index
  float v = u[(size_t)(b * LL + l) * DD + d]
          + u[(size_t)((4 + b) * LL + (LL - 1 - l)) * DD + d]
          + u[(size_t)((8 + b) * LL + lt) * DD + d]
          + u[(size_t)((12 + b) * LL + (LL - 1 - lt)) * DD + d];
  m[i] = v * 0.25f;
}

// ---------------------------------------------------------------------------
// final projection to 3 channels + transpose to (B,C,H,W)
// ---------------------------------------------------------------------------
__global__ void rec2_kernel(const float* __restrict__ t1,
                            const float* __restrict__ w2,
                            const float* __restrict__ b2,
                            float* __restrict__ out) {
  int i = blockIdx.x * blockDim.x + threadIdx.x;    // (b*CIN + c)*LL + l
  if (i >= 4 * CINCH * LL) return;
  int l = i % LL;
  int c = (i / LL) % CINCH;
  int b = i / (LL * CINCH);
  const float* row = t1 + (size_t)(b * LL + l) * DD;
  float acc = b2[c];
  for (int d = 0; d < DD; d++) acc += row[d] * w2[c * DD + d];
  out[i] = acc;
}

// ---------------------------------------------------------------------------
extern "C" void kernel_launch(void* const* d_in, const int* in_sizes, int n_in,
                              void* d_out, int out_size, void* d_ws, size_t ws_size,
                              hipStream_t stream) {
  const float* x         = (const float*)d_in[0];
  const float* patch_w   = (const float*)d_in[1];
  const float* patch_b   = (const float*)d_in[2];
  const float* norm_w    = (const float*)d_in[3];
  const float* in_proj_w = (const float*)d_in[4];
  const float* in_proj_b = (const float*)d_in[5];
  const float* conv_w    = (const float*)d_in[6];
  const float* conv_b    = (const float*)d_in[7];
  const float* dt_bias   = (const float*)d_in[8];
  const float* A_log     = (const float*)d_in[9];
  const float* D_param   = (const float*)d_in[10];
  const float* gnorm_w   = (const float*)d_in[11];
  const float* out_w     = (const float*)d_in[12];
  const float* out_b     = (const float*)d_in[13];
  const float* normf_w   = (const float*)d_in[14];
  const float* rec_w1    = (const float*)d_in[15];
  const float* rec_b1    = (const float*)d_in[16];
  const float* rec_w2    = (const float*)d_in[17];
  const float* rec_b2    = (const float*)d_in[18];

  // workspace carve (floats)
  float* ws = (float*)d_ws;
  size_t off = 0;
  float* hs   = ws + off; off += (size_t)NSEQ * LL * DD;
  float* u    = ws + off; off += (size_t)NSEQ * LL * DD;
  float* proj = ws + off; off += (size_t)NSEQ * LL * PROJC;
  float* xc   = ws + off; off += (size_t)NSEQ * LL * CONVC;
  float* dtb  = ws + off; off += (size_t)NSEQ * LL * NHD;
  float* sb   = ws + off; off += (size_t)NSEQ * NHD * LL;
  float* CBb  = ws + off; off += (size_t)NSEQ * LL * LL;
  float* y    = ws + off; off += (size_t)NSEQ * LL * INTERC;
  float* mbuf = ws + off; off += (size_t)4 * LL * DD;
  float* t1   = ws + off; off += (size_t)4 * LL * DD;
  (void)ws_size; (void)in_sizes; (void)n_in; (void)out_size;

  // patch embed + cross scan
  patch_scan_kernel<<<(NSEQ * LL * DD + 255) / 256, 256, 0, stream>>>(
      x, patch_w, patch_b, hs);

  for (int i = 0; i < NLAY; i++) {
    // u = rmsnorm(hs)
    rmsnorm_kernel<DD><<<NSEQ * LL, DD, 0, stream>>>(hs, norm_w + i * DD, u);

    // proj = u @ ipw.T + ipb   (8192 x 772, K=128)
    {
      int M = NSEQ * LL, Nn = PROJC, Kk = DD;
      int tiles = (M >> 5) * ((Nn + 63) >> 6);
      wmma_gemm_kernel<<<dim3((tiles + 3) / 4, 1), 128, 0, stream>>>(
          u, 0, DD, in_proj_w + (size_t)i * PROJC * DD, 0, DD,
          proj, 0, PROJC, in_proj_b + i * PROJC, M, Nn, Kk, 0);
    }

    // xc = silu(causal_conv(proj[..., 256:768]))
    conv_silu_kernel<<<(NSEQ * LL * CONVC + 255) / 256, 256, 0, stream>>>(
        proj, conv_w + (size_t)i * CONVC * KCONV, conv_b + i * CONVC, xc);

    // dt + cumulative decay
    dtscan_kernel<<<NSEQ * NHD, LL, 0, stream>>>(
        proj, dt_bias + i * NHD, A_log + i * NHD, dtb, sb);

    // CB[s] = Cm @ Bm.T  (512 x 512, K=128, batch=16)
    {
      int tiles = (LL >> 5) * (LL >> 6);
      wmma_gemm_kernel<<<dim3((tiles + 3) / 4, NSEQ), 128, 0, stream>>>(
          xc + INTERC + NSTATE, (long)LL * CONVC, CONVC,
          xc + INTERC,          (long)LL * CONVC, CONVC,
          CBb, (long)LL * LL, LL, nullptr, LL, LL, NSTATE, 0);
    }

    // y = M @ xs + D*xs (causal decay attention), 4 waves/block, 1 t-tile/wave
    ssm_attn_kernel<<<dim3(LL / 64, NSEQ * NHD), 128, 0, stream>>>(
        CBb, sb, dtb, xc, D_param + i * NHD, y);

    // y = rmsnorm(y * silu(gate))
    gated_rmsnorm_kernel<<<NSEQ * LL, INTERC, 0, stream>>>(
        y, proj, gnorm_w + i * INTERC);

    // hs += y @ ow.T + ob   (8192 x 128, K=256)
    {
      int M = NSEQ * LL, Nn = DD, Kk = INTERC;
      int tiles = (M >> 5) * ((Nn + 63) >> 6);
      wmma_gemm_kernel<<<dim3((tiles + 3) / 4, 1), 128, 0, stream>>>(
          y, 0, INTERC, out_w + (size_t)i * DD * INTERC, 0, INTERC,
          hs, 0, DD, out_b + i * DD, M, Nn, Kk, 1);
    }
  }

  // final norm, merge, reconstruction head
  rmsnorm_kernel<DD><<<NSEQ * LL, DD, 0, stream>>>(hs, normf_w, u);
  merge_kernel<<<(4 * LL * DD + 255) / 256, 256, 0, stream>>>(u, mbuf);

  {
    int M = 4 * LL, Nn = DD, Kk = DD;
    int tiles = (M >> 5) * ((Nn + 63) >> 6);
    wmma_gemm_kernel<<<dim3((tiles + 3) / 4, 1), 128, 0, stream>>>(
        mbuf, 0, DD, rec_w1, 0, DD, t1, 0, DD, rec_b1, M, Nn, Kk, 2 /*gelu*/);
  }

  rec2_kernel<<<(4 * CINCH * LL + 255) / 256, 256, 0, stream>>>(
      t1, rec_w2, rec_b2, (float*)d_out);
}